// TypeAttention_85478439125356
// MI455X (gfx1250) — compile-verified
//
#include <hip/hip_runtime.h>
#include <hip/hip_bf16.h>

// ---------------------------------------------------------------------------
// TypeAttention fused pipeline for MI455X (gfx1250, wave32).
//
// Memory-bound: x (361 MB) read ONCE; type_logits (170 MB) written;
// xr (28 MB) written+read. ~0.6 GB total -> ~25 us at 23.3 TB/s HBM.
// fp32 WMMA 16x16x4 keeps reference precision; ~60 GF of compute is far
// below the matrix-op ceiling, so the GEMM is pure data movement.
// ---------------------------------------------------------------------------

#define TOTAL       131072
#define DIM         690
#define KPAD        692        // 4 * 173, zero-padded K for guard-free A frags
#define NUM_CLASSES 53
#define T2          324        // type_num^2
#define NREAL       377        // 324 (liner_w cols) + 53 (relation_weight cols)
#define NUM_BAGS    8192
#define CSTR        388        // padded C-tile row stride (bank skew)

typedef float v2f __attribute__((ext_vector_type(2)));
typedef float v8f __attribute__((ext_vector_type(8)));

// zero pair for branchless tail-step B fragments (lanes with kk == 690)
__device__ const float kZero2[2] = {0.f, 0.f};

// ---------------------------------------------------------------------------
// Kernel 0: TR[i,j] = dot(type_weight[i], relation_weight[j]); 324x53, tiny.
// ---------------------------------------------------------------------------
__global__ __launch_bounds__(256) void tr_kernel(const float* __restrict__ tw,
                                                 const float* __restrict__ rw,
                                                 float* __restrict__ TR) {
  int t = blockIdx.x * 256 + threadIdx.x;
  if (t >= T2 * NUM_CLASSES) return;
  int i = t / NUM_CLASSES;
  int j = t - i * NUM_CLASSES;
  const float* a = tw + (long)i * DIM;
  const float* b = rw + (long)j * DIM;
  float s = 0.f;
  for (int k = 0; k < DIM; ++k) s = fmaf(a[k], b[k], s);
  TR[t] = s;
}

// ---------------------------------------------------------------------------
// Kernel 1: per 16-row tile of x, one workgroup (256 thr = 8 waves):
//   [L | xr] = x_tile @ [liner_w | relation_weight]^T  (fp32 WMMA 16x16x4)
//   type_logits = softmax(L + liner_b)                     -> global output
//   soft_label  = argmax(p + 0.9*max(p)*onehot(type_label))
//   att         = xr[:,query] + TR[soft_label, query]      -> workspace
//   xr          -> workspace (feeds bag logits; x is never re-read)
// ---------------------------------------------------------------------------
__global__ __launch_bounds__(256) void fused_kernel(
    const float* __restrict__ x,
    const float* __restrict__ liner_w,
    const float* __restrict__ liner_b,
    const float* __restrict__ rel_w,
    const int*   __restrict__ type_label,
    const int*   __restrict__ query,
    const float* __restrict__ TR,
    float* __restrict__ type_logits_out,   // [TOTAL, 324]
    float* __restrict__ xr,                // [TOTAL, 53]
    float* __restrict__ att)               // [TOTAL]
{
  // One LDS buffer reused: phase A = x tile (16 x 692 f32 = 44.3 KB),
  // phase B (after barrier) = C tile (16 x 388 f32 = 24.8 KB).
  __shared__ float S[16 * KPAD];

  const int tid     = threadIdx.x;
  const int wave    = tid >> 5;
  const int lane    = tid & 31;
  const int lane_lo = lane & 15;
  const int kq      = (lane >> 4) << 1;   // lanes 0-15 -> K {k0,k0+1}; 16-31 -> {k0+2,k0+3}
  const long m0     = (long)blockIdx.x * 16;

  // ---- stage x tile into LDS with float2 loads (bases are 8B aligned) -----
  for (int i = tid; i < 16 * 345; i += 256) {   // 345 float2 per 690-float row
    const int r = i / 345;
    const int c = (i - r * 345) * 2;
    *(v2f*)&S[r * KPAD + c] = *(const v2f*)(x + (m0 + r) * DIM + c);
  }
  if (tid < 32) S[(tid >> 1) * KPAD + DIM + (tid & 1)] = 0.f;  // K pad -> 0
  __syncthreads();

  // ---- each wave owns 3 N-tiles: {wave, wave+8, wave+16} ------------------
  // Pad columns n in [377,384) feed C columns the epilogue never reads, and
  // GEMM columns are independent, so those lanes just load harmless garbage
  // from a valid address (liner_w row 0) -> fully branchless inner loop.
  const v8f zero8 = {0.f, 0.f, 0.f, 0.f, 0.f, 0.f, 0.f, 0.f};
  v8f acc[3] = {zero8, zero8, zero8};
  const float* brow[3];
#pragma unroll
  for (int t = 0; t < 3; ++t) {
    const int n = (wave + 8 * t) * 16 + lane_lo;
    brow[t] = (n < T2)    ? liner_w + (long)n * DIM
            : (n < NREAL) ? rel_w   + (long)(n - T2) * DIM
                          : liner_w;                      // pad column: unused
  }

  // ---- software-pipelined K loop: 172 full steps, prefetch depth 2 --------
  v2f a_c = *(const v2f*)&S[lane_lo * KPAD + kq];
  v2f b_c[3];
#pragma unroll
  for (int t = 0; t < 3; ++t) b_c[t] = *(const v2f*)(brow[t] + kq);

  for (int k0 = 4; k0 < 688; k0 += 4) {
    const int kk = k0 + kq;
    v2f a_n = *(const v2f*)&S[lane_lo * KPAD + kk];
    v2f b_n[3];
#pragma unroll
    for (int t = 0; t < 3; ++t) b_n[t] = *(const v2f*)(brow[t] + kk);
#pragma unroll
    for (int t = 0; t < 3; ++t)
      acc[t] = __builtin_amdgcn_wmma_f32_16x16x4_f32(
          false, a_c, false, b_c[t], (short)0, acc[t], false, false);
    a_c = a_n;
#pragma unroll
    for (int t = 0; t < 3; ++t) b_c[t] = b_n[t];
  }
  // drain step k0 = 684 (already loaded)
#pragma unroll
  for (int t = 0; t < 3; ++t)
    acc[t] = __builtin_amdgcn_wmma_f32_16x16x4_f32(
        false, a_c, false, b_c[t], (short)0, acc[t], false, false);

  // tail step k0 = 688: kk in {688 (valid pair), 690 (past end)}.
  // Branchless: lanes past the end read from a device-constant zero pair.
  {
    const int kk = 688 + kq;
    v2f a = *(const v2f*)&S[lane_lo * KPAD + kk];        // LDS pad -> zeros
#pragma unroll
    for (int t = 0; t < 3; ++t) {
      const float* bp = (kk + 1 < DIM) ? (brow[t] + kk) : kZero2;
      v2f b = *(const v2f*)bp;
      acc[t] = __builtin_amdgcn_wmma_f32_16x16x4_f32(
          false, a, false, b, (short)0, acc[t], false, false);
    }
  }

  // ---- spill C tiles into LDS (x tile no longer needed) -------------------
  __syncthreads();                                       // x-tile reads done
  const int mbase = (lane >> 4) * 8;                     // VGPR r -> row r or r+8
#pragma unroll
  for (int t = 0; t < 3; ++t) {
    const int n0 = (wave + 8 * t) * 16;
#pragma unroll
    for (int r = 0; r < 8; ++r)
      S[(mbase + r) * CSTR + n0 + lane_lo] = acc[t][r];
  }
  __syncthreads();

  // ---- epilogue: wave w handles rows {w, w+8}; lanes stripe columns -------
  for (int rr = 0; rr < 2; ++rr) {
    const int  m    = wave + rr * 8;
    const long mg   = m0 + m;
    const int  tlab = type_label[mg];
    const int  q    = query[mg];

    float v[11];
    int   cnt = 0;
    float mx = -3.402823e38f;
    for (int c = lane; c < T2; c += 32) {
      float val = S[m * CSTR + c] + liner_b[c];
      v[cnt++] = val;
      mx = fmaxf(mx, val);
    }
    for (int off = 16; off; off >>= 1) mx = fmaxf(mx, __shfl_xor(mx, off, 32));

    // exp, row sum, argmax of nscore. max(softmax) = 1/s, so
    // argmax(p + 0.9*(1/s)*onehot) == argmax(e + 0.9*onehot)  (scale by s>0).
    float sum = 0.f, bestv = -1.f;
    int   besti = 0x7fffffff;
    cnt = 0;
    for (int c = lane; c < T2; c += 32) {
      float e = __expf(v[cnt] - mx);
      v[cnt++] = e;
      sum += e;
      float ns = (c == tlab) ? (e + 0.9f) : e;
      if (ns > bestv) { bestv = ns; besti = c; }          // first-index tie rule
    }
    for (int off = 16; off; off >>= 1) sum += __shfl_xor(sum, off, 32);
    for (int off = 16; off; off >>= 1) {
      float ov = __shfl_xor(bestv, off, 32);
      int   oi = __shfl_xor(besti, off, 32);
      if (ov > bestv || (ov == bestv && oi < besti)) { bestv = ov; besti = oi; }
    }
    const float inv = 1.f / sum;

    // type_logits row (coalesced 128B segments)
    float* outrow = type_logits_out + mg * T2;
    cnt = 0;
    for (int c = lane; c < T2; c += 32) outrow[c] = v[cnt++] * inv;

    // xr row -> workspace
    for (int c = lane; c < NUM_CLASSES; c += 32)
      xr[mg * NUM_CLASSES + c] = S[m * CSTR + T2 + c];

    // attention logit
    if (lane == 0)
      att[mg] = S[m * CSTR + T2 + q] + TR[besti * NUM_CLASSES + q];
  }
}

// ---------------------------------------------------------------------------
// Kernel 2: one wave per bag. Segment softmax over att, then
//   logits[b,c] = sum_i score_i * xr[i,c] + bias[c]
// (identical to segment_sum(score*x) @ RW^T + bias, but reads 53 floats/row
//  instead of 690 -> no second pass over x).
// ---------------------------------------------------------------------------
__global__ __launch_bounds__(256) void bag_kernel(
    const int*   __restrict__ scope,
    const float* __restrict__ att,
    const float* __restrict__ xr,
    const float* __restrict__ bias,
    float* __restrict__ logits)
{
  const int wave = threadIdx.x >> 5;
  const int lane = threadIdx.x & 31;
  const int bag  = blockIdx.x * 8 + wave;
  if (bag >= NUM_BAGS) return;
  const int s = scope[bag], e = scope[bag + 1];

  float mx = -3.402823e38f;
  for (int i = s + lane; i < e; i += 32) mx = fmaxf(mx, att[i]);
  for (int off = 16; off; off >>= 1) mx = fmaxf(mx, __shfl_xor(mx, off, 32));

  float sum = 0.f;
  for (int i = s + lane; i < e; i += 32) sum += __expf(att[i] - mx);
  for (int off = 16; off; off >>= 1) sum += __shfl_xor(sum, off, 32);
  const float inv = 1.f / sum;

  float a0 = 0.f, a1 = 0.f;
  const int c1 = lane + 32;
  for (int i = s; i < e; ++i) {
    const float  w   = __expf(att[i] - mx) * inv;
    const float* row = xr + (long)i * NUM_CLASSES;
    a0 = fmaf(w, row[lane], a0);
    if (c1 < NUM_CLASSES) a1 = fmaf(w, row[c1], a1);
  }
  float* out = logits + (long)bag * NUM_CLASSES;
  out[lane] = a0 + bias[lane];
  if (c1 < NUM_CLASSES) out[c1] = a1 + bias[c1];
}

// ---------------------------------------------------------------------------
extern "C" void kernel_launch(void* const* d_in, const int* in_sizes, int n_in,
                              void* d_out, int out_size, void* d_ws, size_t ws_size,
                              hipStream_t stream) {
  const float* x          = (const float*)d_in[0];
  const int*   scope      = (const int*)  d_in[1];
  const int*   type_label = (const int*)  d_in[2];
  const int*   query      = (const int*)  d_in[3];
  const float* rel_w      = (const float*)d_in[4];
  const float* type_w     = (const float*)d_in[5];
  const float* bias       = (const float*)d_in[6];
  const float* liner_w    = (const float*)d_in[7];
  const float* liner_b    = (const float*)d_in[8];

  float* logits = (float*)d_out;                          // [8192, 53]
  float* tlog   = logits + (long)NUM_BAGS * NUM_CLASSES;  // [131072, 324]

  // workspace layout (~28.4 MB): TR | xr | att
  float* TR  = (float*)d_ws;
  float* xr  = TR + 17408;                                // 324*53 rounded up
  float* att = xr + (long)TOTAL * NUM_CLASSES;

  tr_kernel<<<(T2 * NUM_CLASSES + 255) / 256, 256, 0, stream>>>(type_w, rel_w, TR);
  fused_kernel<<<TOTAL / 16, 256, 0, stream>>>(x, liner_w, liner_b, rel_w,
                                               type_label, query, TR,
                                               tlog, xr, att);
  bag_kernel<<<NUM_BAGS / 8, 256, 0, stream>>>(scope, att, xr, bias, logits);
}